// RoPEELULinearAttention_75728863363110
// MI455X (gfx1250) — compile-verified
//
#include <hip/hip_runtime.h>
#include <math.h>

typedef __attribute__((ext_vector_type(2))) float v2f;
typedef __attribute__((ext_vector_type(8))) float v8f;

#define BATCH 4
#define HEADS 16
#define SEQ   4096
#define DIM   64
#define BHX   (BATCH * HEADS)

// ln(10000)/16
#define ROPE_LC 0.5756462732485115f

__device__ __forceinline__ float elu_p1(float x) {
    // elu(x)+1 : x>0 ? x+1 : exp(x)
    return x > 0.0f ? x + 1.0f : __expf(x);
}

__global__ void zero_ws_kernel(float* __restrict__ p, int n) {
    int i = blockIdx.x * blockDim.x + threadIdx.x;
    if (i < n) p[i] = 0.0f;
}

// Phase 1: per (b,h) chunk of S: Kf = (elu(K)+1)*mask, Kr = rope(Kf),
// partial M[d,e] += Kr^T V via V_WMMA_F32_16X16X4_F32, partial ksum[d] += |Kf|.
__global__ __launch_bounds__(512)
void phase1_kv(const float* __restrict__ K, const float* __restrict__ V,
               const int* __restrict__ mask,
               float* __restrict__ wsM, float* __restrict__ wsKsum) {
    constexpr int STAGE = 64;   // s rows staged in LDS at a time
    constexpr int SPB   = 256;  // s rows per block
    constexpr int LSTR  = 72;   // padded LDS row stride: 2*72 % 64 == 16 (bank-split halves)
    __shared__ float ldsK[STAGE * LSTR];
    __shared__ float ldsV[STAGE * LSTR];

    const int bh     = blockIdx.x / (SEQ / SPB);
    const int sBlock = (blockIdx.x % (SEQ / SPB)) * SPB;
    const int b      = bh / HEADS;
    const int tid    = threadIdx.x;
    const int lane   = tid & 31;
    const int hf     = lane >> 4;   // half-wave select -> K pair {0,1} vs {2,3}
    const int l      = lane & 15;
    const int wave   = tid >> 5;    // 0..15, one 16x16 tile of M each
    const int dT     = (wave >> 2) * 16;
    const int eT     = (wave & 3) * 16;

    const int   p    = tid & 31;    // pair column, invariant across staging loop
    const int   d0   = p * 2;
    const float invf = (p < 16) ? __expf(-ROPE_LC * (float)p) : 0.0f;

    float ks0 = 0.0f, ks1 = 0.0f;
    v8f acc = {};

    const float* Kbase = K + (size_t)bh * SEQ * DIM;
    const float* Vbase = V + (size_t)bh * SEQ * DIM;
    const int*   mbase = mask + b * SEQ;

    for (int stage = 0; stage < SPB / STAGE; ++stage) {
        const int sBase = sBlock + stage * STAGE;
        // ---- load + ELU + mask + RoPE into LDS (pairs of columns) ----
        #pragma unroll
        for (int it = 0; it < (STAGE * 32) / 512; ++it) {
            const int i   = tid + it * 512;
            const int row = i >> 5;
            const int s   = sBase + row;
            const v2f kv = *(const v2f*)(Kbase + (size_t)s * DIM + d0);
            const v2f vv = *(const v2f*)(Vbase + (size_t)s * DIM + d0);
            const float m  = (float)mbase[s];
            const float k0 = elu_p1(kv.x) * m;
            const float k1 = elu_p1(kv.y) * m;
            ks0 += fabsf(k0);
            ks1 += fabsf(k1);
            float r0, r1;
            if (p < 16) {  // rotary half
                float sn, cs;
                sincosf((float)s * invf, &sn, &cs);
                r0 = k0 * cs - k1 * sn;
                r1 = k1 * cs + k0 * sn;
            } else {       // pass-through half
                r0 = k0; r1 = k1;
            }
            ldsK[row * LSTR + d0]     = r0;
            ldsK[row * LSTR + d0 + 1] = r1;
            ldsV[row * LSTR + d0]     = vv.x;
            ldsV[row * LSTR + d0 + 1] = vv.y;
        }
        __syncthreads();
        // ---- M[d,e] += Kr^T V  (contract over s, K=4 per WMMA) ----
        #pragma unroll 4
        for (int sl = 0; sl < STAGE; sl += 4) {
            const int rA = sl + 2 * hf;
            v2f a, bb;
            a.x  = ldsK[rA * LSTR + dT + l];        // A: M=d (all lanes), K=s
            a.y  = ldsK[(rA + 1) * LSTR + dT + l];
            bb.x = ldsV[rA * LSTR + eT + l];        // B: K=s, N=e
            bb.y = ldsV[(rA + 1) * LSTR + eT + l];
            acc = __builtin_amdgcn_wmma_f32_16x16x4_f32(false, a, false, bb,
                                                        (short)0, acc, false, false);
        }
        __syncthreads();
    }

    // flush this wave's partial 16x16 tile of M
    float* Mh = wsM + (size_t)bh * DIM * DIM;
    #pragma unroll
    for (int r = 0; r < 8; ++r) {
        const int d = dT + r + 8 * hf;              // C layout: M = r (+8 for high lanes)
        atomicAdd(&Mh[d * DIM + eT + l], acc[r]);
    }
    atomicAdd(&wsKsum[bh * DIM + d0],     ks0);
    atomicAdd(&wsKsum[bh * DIM + d0 + 1], ks1);
}

// Phase 2: out[s,e] = (rope(Qf) @ M)[s,e] / (Qf[s,:] . ksum)
__global__ __launch_bounds__(256)
void phase2_q(const float* __restrict__ Q, const float* __restrict__ wsM,
              const float* __restrict__ wsKsum, float* __restrict__ out) {
    constexpr int LSTR = 72;
    constexpr int ROWS = 32;     // s rows per block
    __shared__ float ldsM[DIM * LSTR];
    __shared__ float ldsQ[ROWS * LSTR];
    __shared__ float ldsN[ROWS];

    const int bh     = blockIdx.x >> 7;                 // SEQ/ROWS == 128
    const int sBlock = (blockIdx.x & 127) * ROWS;
    const int tid    = threadIdx.x;
    const int lane   = tid & 31;
    const int hf     = lane >> 4;
    const int l      = lane & 15;
    const int wave   = tid >> 5;                        // 0..7
    const int sT     = (wave >> 2) * 16;
    const int eT     = (wave & 3) * 16;

    // stage M (64x64) into LDS; L2-resident across the 128 blocks of this head
    const float* Mh = wsM + (size_t)bh * DIM * DIM;
    for (int i = tid; i < DIM * DIM; i += 256)
        ldsM[(i >> 6) * LSTR + (i & 63)] = Mh[i];
    if (tid < ROWS) ldsN[tid] = 0.0f;
    __syncthreads();

    const int   p    = tid & 31;
    const int   d0   = p * 2;
    const float invf = (p < 16) ? __expf(-ROPE_LC * (float)p) : 0.0f;
    const float ksa  = wsKsum[bh * DIM + d0];
    const float ksb  = wsKsum[bh * DIM + d0 + 1];
    const float* Qbase = Q + (size_t)bh * SEQ * DIM;

    #pragma unroll
    for (int it = 0; it < (ROWS * 32) / 256; ++it) {
        const int i   = tid + it * 256;
        const int row = i >> 5;
        const int s   = sBlock + row;
        const v2f q  = *(const v2f*)(Qbase + (size_t)s * DIM + d0);
        const float q0 = elu_p1(q.x * 0.125f);   // scaling = D^-0.5
        const float q1 = elu_p1(q.y * 0.125f);
        atomicAdd(&ldsN[row], q0 * ksa + q1 * ksb);   // norm = Qf . ksum (pre-rope)
        float r0, r1;
        if (p < 16) {
            float sn, cs;
            sincosf((float)s * invf, &sn, &cs);
            r0 = q0 * cs - q1 * sn;
            r1 = q1 * cs + q0 * sn;
        } else {
            r0 = q0; r1 = q1;
        }
        ldsQ[row * LSTR + d0]     = r0;
        ldsQ[row * LSTR + d0 + 1] = r1;
    }
    __syncthreads();

    // out tile = Qr(16xK) @ M(Kx16), contract over d in steps of 4
    v8f acc = {};
    #pragma unroll
    for (int k0 = 0; k0 < DIM; k0 += 4) {
        v2f a = *(const v2f*)&ldsQ[(sT + l) * LSTR + k0 + 2 * hf];  // A: M=s, K=d
        v2f bb;
        bb.x = ldsM[(k0 + 2 * hf) * LSTR + eT + l];                 // B: K=d, N=e
        bb.y = ldsM[(k0 + 2 * hf + 1) * LSTR + eT + l];
        acc = __builtin_amdgcn_wmma_f32_16x16x4_f32(false, a, false, bb,
                                                    (short)0, acc, false, false);
    }

    float* obase = out + ((size_t)bh * SEQ + sBlock) * DIM;
    #pragma unroll
    for (int r = 0; r < 8; ++r) {
        const int row = sT + r + 8 * hf;
        obase[(size_t)row * DIM + eT + l] = acc[r] / ldsN[row];
    }
}

extern "C" void kernel_launch(void* const* d_in, const int* in_sizes, int n_in,
                              void* d_out, int out_size, void* d_ws, size_t ws_size,
                              hipStream_t stream) {
    (void)in_sizes; (void)n_in; (void)out_size; (void)ws_size;
    const float* Q    = (const float*)d_in[0];
    const float* K    = (const float*)d_in[1];
    const float* V    = (const float*)d_in[2];
    const int*   mask = (const int*)d_in[3];
    float* out    = (float*)d_out;
    float* wsM    = (float*)d_ws;                        // BHX * 64 * 64 floats
    float* wsKsum = wsM + (size_t)BHX * DIM * DIM;       // BHX * 64 floats

    const int nz = BHX * DIM * DIM + BHX * DIM;
    zero_ws_kernel<<<(nz + 255) / 256, 256, 0, stream>>>(wsM, nz);
    phase1_kv<<<BHX * (SEQ / 256), 512, 0, stream>>>(K, V, mask, wsM, wsKsum);
    phase2_q<<<BHX * (SEQ / 32), 256, 0, stream>>>(Q, wsM, wsKsum, out);
}